// SAGE_DGL_63110249447723
// MI455X (gfx1250) — compile-verified
//
#include <hip/hip_runtime.h>

typedef float v2f __attribute__((ext_vector_type(2)));
typedef float v8f __attribute__((ext_vector_type(8)));

#define D_IN  128
#define HID   256
#define OUTD  64
#define SIZE1 25000
#define SIZE2 5000

__device__ __forceinline__ v8f wmma_f32x4(v2f a, v2f b, v8f c) {
    // D = A(16x4 f32) * B(4x16 f32) + C(16x16 f32)
    return __builtin_amdgcn_wmma_f32_16x16x4_f32(
        /*neg_a=*/false, a, /*neg_b=*/false, b,
        /*c_mod=*/(short)0, c, /*reuse_a=*/false, /*reuse_b=*/false);
}

// One wave per edge: gather src row (L2-resident), atomic-add into summed[dst].
template<int D>
__global__ void sage_aggregate(const float* __restrict__ x,
                               const int* __restrict__ src,
                               const int* __restrict__ dst,
                               float* __restrict__ summed,
                               float* __restrict__ deg,
                               int n_edges) {
    const int wave = blockIdx.x * (blockDim.x >> 5) + (threadIdx.x >> 5);
    const int lane = threadIdx.x & 31;
    if (wave >= n_edges) return;                      // wave-uniform branch
    const int s = src[wave];
    const int t = dst[wave];
    const float* __restrict__ xs = x + (size_t)s * D;
    float* __restrict__ out = summed + (size_t)t * D;
#pragma unroll
    for (int c = lane * 4; c < D; c += 128) {
        const float4 v = *reinterpret_cast<const float4*>(xs + c);
        atomicAdd(out + c + 0, v.x);
        atomicAdd(out + c + 1, v.y);
        atomicAdd(out + c + 2, v.z);
        atomicAdd(out + c + 3, v.w);
    }
    if (lane == 0) atomicAdd(deg + t, 1.0f);
}

// Fused: out[m,n] = act( xdst[m,:]@Wself + (summed[m,:]/max(deg,1))@Wneigh + bias )
// One wave owns a 16x(16*TILES) strip: a single A load feeds TILES WMMAs.
// Block has N/(16*TILES) waves; grid.x covers M in 16-row tiles.
template<int K, int N, int TILES, bool RELU>
__global__ void sage_gemm(const float* __restrict__ xdst,   // [M,K]
                          const float* __restrict__ summed, // [M,K]
                          const float* __restrict__ deg,    // [M]
                          const float* __restrict__ Wself,  // [K,N]
                          const float* __restrict__ Wneigh, // [K,N]
                          const float* __restrict__ bias,   // [N]
                          float* __restrict__ out,          // [M,N]
                          int M) {
    const int lane = threadIdx.x & 31;
    const int wv   = threadIdx.x >> 5;
    const int row0 = blockIdx.x * 16;
    const int col0 = wv * (16 * TILES);
    const int mrow = lane & 15;               // A-matrix M row held by this lane
    const int koff = (lane >> 4) << 1;        // lanes 0-15: K0/K1, lanes 16-31: K2/K3
    const int ncol = col0 + (lane & 15);      // base N column held by this lane

    int r = row0 + mrow;
    if (r >= M) r = M - 1;                    // clamp tail reads; stores guarded below
    const float invd = 1.0f / fmaxf(deg[r], 1.0f);
    const float* __restrict__ xr = xdst   + (size_t)r * K;
    const float* __restrict__ sr = summed + (size_t)r * K;

    v8f acc[TILES];
#pragma unroll
    for (int t = 0; t < TILES; ++t) acc[t] = v8f{0.f,0.f,0.f,0.f,0.f,0.f,0.f,0.f};

#pragma unroll 4
    for (int k = 0; k < K; k += 4) {          // self term
        v2f a;
        a.x = xr[k + koff];
        a.y = xr[k + koff + 1];
        const float* w0 = Wself + (size_t)(k + koff) * N + ncol;
#pragma unroll
        for (int t = 0; t < TILES; ++t) {
            v2f b;
            b.x = w0[16 * t];
            b.y = w0[N + 16 * t];
            acc[t] = wmma_f32x4(a, b, acc[t]);
        }
    }
#pragma unroll 4
    for (int k = 0; k < K; k += 4) {          // neighbor-mean term
        v2f a;
        a.x = sr[k + koff]     * invd;
        a.y = sr[k + koff + 1] * invd;
        const float* w0 = Wneigh + (size_t)(k + koff) * N + ncol;
#pragma unroll
        for (int t = 0; t < TILES; ++t) {
            v2f b;
            b.x = w0[16 * t];
            b.y = w0[N + 16 * t];
            acc[t] = wmma_f32x4(a, b, acc[t]);
        }
    }

    const int mbase = row0 + ((lane >> 4) << 3);  // lanes 16-31 hold M rows +8
#pragma unroll
    for (int t = 0; t < TILES; ++t) {
        const float bv = bias[ncol + 16 * t];
#pragma unroll
        for (int j = 0; j < 8; ++j) {
            const int m = mbase + j;
            if (m < M) {
                float v = acc[t][j] + bv;
                if (RELU) v = fmaxf(v, 0.0f);
                out[(size_t)m * N + ncol + 16 * t] = v;
            }
        }
    }
}

// One wave per row of 64 logits; in-place log_softmax via lane shuffles.
__global__ void log_softmax64(float* __restrict__ io, int M) {
    const int row  = blockIdx.x * (blockDim.x >> 5) + (threadIdx.x >> 5);
    const int lane = threadIdx.x & 31;
    if (row >= M) return;
    float* __restrict__ p = io + (size_t)row * 64;
    const float v0 = p[lane];
    const float v1 = p[lane + 32];
    float mx = fmaxf(v0, v1);
#pragma unroll
    for (int off = 16; off; off >>= 1) mx = fmaxf(mx, __shfl_xor(mx, off, 32));
    float s = expf(v0 - mx) + expf(v1 - mx);
#pragma unroll
    for (int off = 16; off; off >>= 1) s += __shfl_xor(s, off, 32);
    const float lse = mx + logf(s);
    p[lane]      = v0 - lse;
    p[lane + 32] = v1 - lse;
}

extern "C" void kernel_launch(void* const* d_in, const int* in_sizes, int n_in,
                              void* d_out, int out_size, void* d_ws, size_t ws_size,
                              hipStream_t stream) {
    (void)n_in; (void)out_size; (void)ws_size;

    const float* x       = (const float*)d_in[0];
    const float* Wself1  = (const float*)d_in[1];
    const float* Wneigh1 = (const float*)d_in[2];
    const float* b1      = (const float*)d_in[3];
    const float* Wself2  = (const float*)d_in[4];
    const float* Wneigh2 = (const float*)d_in[5];
    const float* b2      = (const float*)d_in[6];
    const int*   esrc1   = (const int*)d_in[7];
    const int*   edst1   = (const int*)d_in[8];
    const int*   esrc2   = (const int*)d_in[9];
    const int*   edst2   = (const int*)d_in[10];
    const int E1 = in_sizes[7];
    const int E2 = in_sizes[9];

    // Workspace layout (bytes, 256-aligned):
    char* ws = (char*)d_ws;
    float* summed1 = (float*)(ws + 0);          // 25000*128*4 = 12,800,000
    float* deg1    = (float*)(ws + 12800000);   // 25000*4 (padded to 102,400)
    float* h       = (float*)(ws + 12902400);   // 25000*256*4 = 25,600,000
    float* summed2 = (float*)(ws + 38502400);   // 5000*256*4 = 5,120,000
    float* deg2    = (float*)(ws + 43622400);   // 5000*4 = 20,000
    float* logits  = (float*)d_out;             // 5000*64

    hipMemsetAsync(summed1, 0, 12902400, stream);   // summed1 + deg1
    hipMemsetAsync(summed2, 0, 5140000,  stream);   // summed2 + deg2

    // Layer 1: aggregate  (wave per edge, 8 waves/block)
    {
        dim3 g((E1 + 7) / 8), b(256);
        sage_aggregate<D_IN><<<g, b, 0, stream>>>(x, esrc1, edst1, summed1, deg1, E1);
    }
    // Layer 1: fused mean + GEMM + bias + ReLU -> h [25000,256]
    // 4 waves/block, each wave 16x64 strip => block covers 16x256.
    {
        dim3 g((SIZE1 + 15) / 16), b(128);
        sage_gemm<D_IN, HID, 4, true><<<g, b, 0, stream>>>(
            x, summed1, deg1, Wself1, Wneigh1, b1, h, SIZE1);
    }
    // Layer 2: aggregate over h
    {
        dim3 g((E2 + 7) / 8), b(256);
        sage_aggregate<HID><<<g, b, 0, stream>>>(h, esrc2, edst2, summed2, deg2, E2);
    }
    // Layer 2: fused mean + GEMM + bias -> logits [5000,64]
    // 1 wave/block covers all 64 columns.
    {
        dim3 g((SIZE2 + 15) / 16), b(32);
        sage_gemm<HID, OUTD, 4, false><<<g, b, 0, stream>>>(
            h, summed2, deg2, Wself2, Wneigh2, b2, logits, SIZE2);
    }
    // log_softmax rows (in-place on d_out)
    {
        dim3 g((SIZE2 + 7) / 8), b(256);
        log_softmax64<<<g, b, 0, stream>>>(logits, SIZE2);
    }
}